// Histogram_Binning_33818572488971
// MI455X (gfx1250) — compile-verified
//
#include <hip/hip_runtime.h>

#define NBINS 15
#define NCLS  19
#define HW_SHIFT 18               // 512*512 = 2^18
#define HW (1 << HW_SHIFT)
#define LOG2E 1.4426950408889634f

typedef float f32x4 __attribute__((ext_vector_type(4)));
typedef float f32x2 __attribute__((ext_vector_type(2)));
typedef __attribute__((address_space(1))) int as1_int;
typedef __attribute__((address_space(3))) int as3_int;

__global__ __launch_bounds__(256)
void calib_hist_kernel(const float* __restrict__ logits,
                       const float* __restrict__ val_freqs,
                       float* __restrict__ out, int nvec)
{
    // Stage the 19x15 calibration table into LDS via the CDNA5 async
    // global->LDS path (ASYNCcnt + s_wait_asynccnt).
    __shared__ float s_vf[NCLS * NBINS];
    for (int i = threadIdx.x; i < NCLS * NBINS; i += 256) {
#if __has_builtin(__builtin_amdgcn_global_load_async_to_lds_b32)
        __builtin_amdgcn_global_load_async_to_lds_b32(
            (as1_int*)(val_freqs + i), (as3_int*)(&s_vf[i]), 0, 0);
#else
        s_vf[i] = val_freqs[i];
#endif
    }
#if __has_builtin(__builtin_amdgcn_global_load_async_to_lds_b32)
# if __has_builtin(__builtin_amdgcn_s_wait_asynccnt)
    __builtin_amdgcn_s_wait_asynccnt(0);
# else
    asm volatile("s_wait_asynccnt 0" ::: "memory");
# endif
#endif
    __syncthreads();

    const int g = blockIdx.x * 256 + threadIdx.x;   // vec4-pixel group id
    if (g >= nvec) return;

    const int p0   = g << 2;                        // first pixel (b*HW + sp)
    const int b    = p0 >> HW_SHIFT;                // batch index
    const int base = p0 + ((b * (NCLS - 1)) << HW_SHIFT); // elem(c)=base+c*HW

    // 19 channels x 4 pixels via b128 non-temporal loads, split into two
    // float2 halves so the bulk arithmetic maps to v_pk_*_f32 (VOP3P).
    f32x2 xh[2][NCLS];
#pragma unroll
    for (int c = 0; c < NCLS; ++c) {
        const f32x4 v = __builtin_nontemporal_load(
            (const f32x4*)(logits + base + c * HW));
        xh[0][c].x = v.x; xh[0][c].y = v.y;
        xh[1][c].x = v.z; xh[1][c].y = v.w;
    }

#pragma unroll
    for (int h = 0; h < 2; ++h) {
        f32x2* x = xh[h];

        // softmax max (scalar max pairs; VOPD-fusable)
        f32x2 m = x[0];
#pragma unroll
        for (int c = 1; c < NCLS; ++c) m = __builtin_elementwise_max(m, x[c]);
        const f32x2 mb = m * LOG2E;

        // e = exp2(x*log2e - m*log2e): v_pk_fma_f32 + 2x v_exp_f32
        f32x2 se = {0.0f, 0.0f};
#pragma unroll
        for (int c = 0; c < NCLS; ++c) {
            const f32x2 a = x[c] * LOG2E - mb;      // v_pk_fma_f32
            f32x2 e;
            e.x = __builtin_amdgcn_exp2f(a.x);
            e.y = __builtin_amdgcn_exp2f(a.y);
            x[c] = e;
            se += e;                                 // v_pk_add_f32
        }

        // bin = clip((int)(p*15),0,14); fold p=e/se into t = 15*rcp(se)
        f32x2 t;
        t.x = 15.0f * __builtin_amdgcn_rcpf(se.x);
        t.y = 15.0f * __builtin_amdgcn_rcpf(se.y);

        f32x2 s = {0.0f, 0.0f};
#pragma unroll
        for (int c = 0; c < NCLS; ++c) {
            const f32x2 p = x[c] * t;               // v_pk_mul_f32
            int b0 = (int)p.x, b1 = (int)p.y;
            b0 = b0 < 0 ? 0 : (b0 > NBINS - 1 ? NBINS - 1 : b0);
            b1 = b1 < 0 ? 0 : (b1 > NBINS - 1 ? NBINS - 1 : b1);
            f32x2 cal;
            cal.x = s_vf[c * NBINS + b0];           // ds_load_b32 gather
            cal.y = s_vf[c * NBINS + b1];
            x[c] = cal;
            s += cal;                                // v_pk_add_f32
        }

        // renormalize over classes (in place; keeps VGPR pressure ~100)
        f32x2 inv;
        inv.x = __builtin_amdgcn_rcpf(s.x == 0.0f ? 1.0f : s.x);
        inv.y = __builtin_amdgcn_rcpf(s.y == 0.0f ? 1.0f : s.y);
#pragma unroll
        for (int c = 0; c < NCLS; ++c) x[c] *= inv; // v_pk_mul_f32
    }

    // b128 non-temporal streaming stores (output never re-read)
#pragma unroll
    for (int c = 0; c < NCLS; ++c) {
        f32x4 v;
        v.x = xh[0][c].x; v.y = xh[0][c].y;
        v.z = xh[1][c].x; v.w = xh[1][c].y;
        __builtin_nontemporal_store(v, (f32x4*)(out + base + c * HW));
    }
}

extern "C" void kernel_launch(void* const* d_in, const int* in_sizes, int n_in,
                              void* d_out, int out_size, void* d_ws, size_t ws_size,
                              hipStream_t stream) {
    const float* logits    = (const float*)d_in[0];   // [8,19,512,512] f32
    const float* val_freqs = (const float*)d_in[1];   // [19,15] f32
    float*       out       = (float*)d_out;           // [8,19,512,512] f32

    const int npix = in_sizes[0] / NCLS;   // B*H*W = 2,097,152
    const int nvec = npix >> 2;            // float4 groups = 524,288
    const int blocks = (nvec + 255) / 256; // 2048 blocks of 8 waves

    calib_hist_kernel<<<blocks, 256, 0, stream>>>(logits, val_freqs, out, nvec);
}